// MemoryTimeUnit_19301583028277
// MI455X (gfx1250) — compile-verified
//
#include <hip/hip_runtime.h>
#include <hip/hip_bf16.h>
#include <stddef.h>

// Problem constants (from reference)
#define BB 8      // batch
#define PP 64     // prefix / output time steps
#define VV 8      // variates
#define LPTS 1024 // ts length
#define DD 256    // channels
#define NSEQ (BB*VV)        // 64 independent sequences
#define NCHUNK 8            // chunks for the backward suffix scan
#define CHUNK (LPTS/NCHUNK) // 128

typedef __attribute__((ext_vector_type(2))) float v2f;
typedef __attribute__((ext_vector_type(8))) float v8f;

// ---------------------------------------------------------------------------
// Stage 1: chunked reverse scan of the ts part for the backward LRU.
// For sequence n (b = n/V, v = n%V), chunk c, channel d:
//   T[n][c][d] = sum_{j=0}^{127} lam_b^j * (ts[b, c*128+j, v, d] + signal[d])
// Grid: 512 blocks (n*8+c), 256 threads (d). Coalesced 128B/wave loads,
// streaming reverse through HBM; this stage moves the 67MB that bounds the op.
// ---------------------------------------------------------------------------
__global__ void lru_stage1(const float* __restrict__ ts,
                           const float* __restrict__ sig,
                           const float* __restrict__ bnu,
                           const float* __restrict__ bth,
                           float* __restrict__ partials) {
  const int d = threadIdx.x;
  const int n = blockIdx.x / NCHUNK;
  const int c = blockIdx.x % NCHUNK;
  const int b = n / VV, v = n % VV;

  const float r  = __expf(-__expf(bnu[d]));
  const float th = bth[d];
  const float lr = r * __cosf(th);
  const float li = r * __sinf(th);
  const float s  = sig[d];

  float Tr = 0.f, Ti = 0.f;
  // start at the last row of the chunk, walk backwards (stride V*D floats)
  const float* p = ts + (((size_t)b * LPTS + (size_t)(c * CHUNK + CHUNK - 1)) * VV + v) * DD + d;
  const int stride = VV * DD;

  #pragma unroll 4
  for (int j = CHUNK - 1; j >= 0; --j) {
    if (j >= 8) __builtin_prefetch(p - 8 * stride, 0, 1);  // global_prefetch ahead of stream
    const float x = *p + s;
    const float nr = lr * Tr - li * Ti + x;
    const float ni = lr * Ti + li * Tr;
    Tr = nr; Ti = ni;
    p -= stride;
  }

  const size_t idx = ((size_t)(n * NCHUNK + c) * DD + d) * 2;
  partials[idx]     = Tr;
  partials[idx + 1] = Ti;
}

// ---------------------------------------------------------------------------
// Stage 2: combine chunk partials (Horner with lam_b^128), then run the
// 64-step forward scan and 64-step backward tail scan over the prefix.
// Writes Y[m=n*64+t, 0:256]=y_fwd, Y[m, 256:512]=y_bwd (GEMM operand layout).
// Grid: 64 blocks (n), 256 threads (d).
// ---------------------------------------------------------------------------
__global__ void lru_stage2(const float* __restrict__ mem,
                           const float* __restrict__ pre,
                           const float* __restrict__ fnu, const float* __restrict__ fth,
                           const float* __restrict__ fgr, const float* __restrict__ fgi,
                           const float* __restrict__ bnu, const float* __restrict__ bth,
                           const float* __restrict__ bgr, const float* __restrict__ bgi,
                           const float* __restrict__ partials,
                           float* __restrict__ Y) {
  const int d = threadIdx.x;
  const int n = blockIdx.x;
  const int b = n / VV;

  // lam_b
  const float rb  = __expf(-__expf(bnu[d]));
  const float thb = bth[d];
  const float lbr = rb * __cosf(thb);
  const float lbi = rb * __sinf(thb);

  // lam_b^128 by 7 complex squarings (accurate, avoids large-angle trig)
  float qr = lbr, qi = lbi;
  #pragma unroll
  for (int k = 0; k < 7; ++k) {
    const float nr = qr * qr - qi * qi;
    const float ni = 2.f * qr * qi;
    qr = nr; qi = ni;
  }

  // S = hb[P] = sum_c lam_b^{128c} * T_c   (Horner, c = 7..0)
  float Sr = 0.f, Si = 0.f;
  #pragma unroll
  for (int c = NCHUNK - 1; c >= 0; --c) {
    const size_t idx = ((size_t)(n * NCHUNK + c) * DD + d) * 2;
    const float nr = Sr * qr - Si * qi + partials[idx];
    const float ni = Sr * qi + Si * qr + partials[idx + 1];
    Sr = nr; Si = ni;
  }

  // lam_f and output gains
  const float rf  = __expf(-__expf(fnu[d]));
  const float thf = fth[d];
  const float lfr = rf * __cosf(thf);
  const float lfi = rf * __sinf(thf);
  const float gfr = fgr[d], gfi = fgi[d];
  const float gbr = bgr[d], gbi = bgi[d];
  const float pe  = pre[d];

  // Forward scan over the prefix: y_fwd[t] = Re(g_f * h[t])
  float hr = 0.f, hi = 0.f;
  for (int t = 0; t < PP; ++t) {
    const float x = mem[((size_t)b * PP + t) * DD + d] + pe;
    const float nr = lfr * hr - lfi * hi + x;
    const float ni = lfr * hi + lfi * hr;
    hr = nr; hi = ni;
    Y[(size_t)(n * PP + t) * (2 * DD) + d] = gfr * hr - gfi * hi;
  }

  // Backward tail: hb[t] = lam_b*hb[t+1] + Z[t], seeded with hb[P] = S
  float ur = Sr, ui = Si;
  for (int t = PP - 1; t >= 0; --t) {
    const float x = mem[((size_t)b * PP + t) * DD + d] + pe;
    const float nr = lbr * ur - lbi * ui + x;
    const float ni = lbr * ui + lbi * ur;
    ur = nr; ui = ni;
    Y[(size_t)(n * PP + t) * (2 * DD) + DD + d] = gbr * ur - gbi * ui;
  }
}

// ---------------------------------------------------------------------------
// Stage 3: projection GEMM with native f32 WMMA, 2x2 register blocking.
//   Zp[m, i] = sum_k Y[m, k] * W[i, k],  M=4096, K=512, N=256
// One wave computes a 32x32 output patch = 2x2 tiles of 16x16: per K=4 step
// it loads 2 A frags + 2 B frags (4 x b64) and issues 4 WMMAs (1:1 ratio,
// 2x less Y/W re-read traffic; everything stays L2-resident).
// A 16x4 layout (ISA 7.12.2): lanes 0-15 -> M=0..15; VGPR0 = K{0|2}, VGPR1 =
// K{1|3} by lane-half. B 4x16 mirrored. D 16x16: VGPR r -> M = r + 8*(lane>=16),
// N = lane&15.
// ---------------------------------------------------------------------------
__global__ void proj_wmma(const float* __restrict__ Y,
                          const float* __restrict__ W,
                          float* __restrict__ Zp) {
  const int lane = threadIdx.x & 31;
  const int wave = threadIdx.x >> 5;
  const int w    = blockIdx.x * 8 + wave;   // 1024 waves total
  const int mp = w >> 3;                    // 128 m-pairs  (32 rows each)
  const int ip = w & 7;                     // 8   i-pairs  (32 cols each)
  const int m0 = mp * 32, i0 = ip * 32;
  const int row = lane & 15;                // A row / B,N column
  const int hi  = lane >> 4;                // lane half -> K pair select

  const float* ya0 = Y + (size_t)(m0 + row) * (2 * DD) + 2 * hi;
  const float* ya1 = ya0 + (size_t)16 * (2 * DD);
  const float* wb0 = W + (size_t)(i0 + row) * (2 * DD) + 2 * hi;
  const float* wb1 = wb0 + (size_t)16 * (2 * DD);

  v8f acc00 = {}, acc01 = {}, acc10 = {}, acc11 = {};
  #pragma unroll 4
  for (int k = 0; k < 2 * DD; k += 4) {
    const v2f a0 = *(const v2f*)(ya0 + k);
    const v2f a1 = *(const v2f*)(ya1 + k);
    const v2f b0 = *(const v2f*)(wb0 + k);
    const v2f b1 = *(const v2f*)(wb1 + k);
    // 8 args: (neg_a, A, neg_b, B, c_mod, C, reuse_a, reuse_b)
    acc00 = __builtin_amdgcn_wmma_f32_16x16x4_f32(false, a0, false, b0,
                                                  (short)0, acc00, false, false);
    acc01 = __builtin_amdgcn_wmma_f32_16x16x4_f32(false, a0, false, b1,
                                                  (short)0, acc01, false, false);
    acc10 = __builtin_amdgcn_wmma_f32_16x16x4_f32(false, a1, false, b0,
                                                  (short)0, acc10, false, false);
    acc11 = __builtin_amdgcn_wmma_f32_16x16x4_f32(false, a1, false, b1,
                                                  (short)0, acc11, false, false);
  }

  #pragma unroll
  for (int r = 0; r < 8; ++r) {
    const size_t rm0 = (size_t)(m0 + r + 8 * hi) * DD;
    const size_t rm1 = (size_t)(m0 + 16 + r + 8 * hi) * DD;
    Zp[rm0 + i0 + row]      = acc00[r];
    Zp[rm0 + i0 + 16 + row] = acc01[r];
    Zp[rm1 + i0 + row]      = acc10[r];
    Zp[rm1 + i0 + 16 + row] = acc11[r];
  }
}

// ---------------------------------------------------------------------------
// Stage 4: mean over V + bias.  out[b, p, i] = mean_v Zp[(b*V+v)*P + p, i] + b[i]
// ---------------------------------------------------------------------------
__global__ void mean_bias(const float* __restrict__ Zp,
                          const float* __restrict__ bias,
                          float* __restrict__ out) {
  const int idx = blockIdx.x * 256 + threadIdx.x;  // 131072 outputs
  const int i = idx & 255;
  const int p = (idx >> 8) & 63;
  const int b = idx >> 14;
  float s = 0.f;
  #pragma unroll
  for (int v = 0; v < VV; ++v)
    s += Zp[(size_t)((b * VV + v) * PP + p) * DD + i];
  out[idx] = s * (1.f / VV) + bias[i];
}

// ---------------------------------------------------------------------------
extern "C" void kernel_launch(void* const* d_in, const int* in_sizes, int n_in,
                              void* d_out, int out_size, void* d_ws, size_t ws_size,
                              hipStream_t stream) {
  const float* memory = (const float*)d_in[0];
  const float* ts     = (const float*)d_in[1];
  const float* fnu    = (const float*)d_in[2];
  const float* fth    = (const float*)d_in[3];
  const float* fgr    = (const float*)d_in[4];
  const float* fgi    = (const float*)d_in[5];
  const float* bnu    = (const float*)d_in[6];
  const float* bth    = (const float*)d_in[7];
  const float* bgr    = (const float*)d_in[8];
  const float* bgi    = (const float*)d_in[9];
  const float* projW  = (const float*)d_in[10];
  const float* projB  = (const float*)d_in[11];
  const float* preEmb = (const float*)d_in[12];
  const float* sigEmb = (const float*)d_in[13];
  float* out = (float*)d_out;

  // Workspace layout (floats):
  //   partials: NSEQ*NCHUNK*DD*2          = 262144
  //   Y:        (NSEQ*PP) * (2*DD)        = 2097152
  //   Zp:       (NSEQ*PP) * DD            = 1048576
  float* ws       = (float*)d_ws;
  float* partials = ws;
  float* Yb       = ws + (size_t)NSEQ * NCHUNK * DD * 2;
  float* Zp       = Yb + (size_t)NSEQ * PP * 2 * DD;

  lru_stage1<<<NSEQ * NCHUNK, DD, 0, stream>>>(ts, sigEmb, bnu, bth, partials);
  lru_stage2<<<NSEQ, DD, 0, stream>>>(memory, preEmb, fnu, fth, fgr, fgi,
                                      bnu, bth, bgr, bgi, partials, Yb);
  // 4096 output tiles / 4 tiles per wave / 8 waves per block = 128 blocks
  proj_wmma<<<128, 256, 0, stream>>>(Yb, projW, Zp);
  mean_bias<<<(BB * PP * DD) / 256, 256, 0, stream>>>(Zp, projB, out);
}